// SparseConv3d_79714593013864
// MI455X (gfx1250) — compile-verified
//
#include <hip/hip_runtime.h>

typedef float v2f __attribute__((ext_vector_type(2)));
typedef float v8f __attribute__((ext_vector_type(8)));

#define BB   2
#define DD   128
#define HH   128
#define WW   128
#define CI   64
#define CO   64
#define NOFF 27            // 3x3x3 taps
#define NPTS 200000
#define GRID_CELLS (BB*DD*HH*WW)          // 4,194,304
#define WT_FLOATS  (NOFF*16*4*32*2)       // 110,592 floats = 442,368 B
#define A_STRIDE   68                     // 64 + 4 pad -> conflict-free ds_load_b64

// ---------------------------------------------------------------- grid build
__global__ void k_fill_grid(int* __restrict__ grid) {
    int i = blockIdx.x * blockDim.x + threadIdx.x;
    if (i < GRID_CELLS) grid[i] = -1;
}

__global__ void k_scatter(const int* __restrict__ coords, int* __restrict__ grid, int n) {
    int i = blockIdx.x * blockDim.x + threadIdx.x;
    if (i >= n) return;
    int b = coords[i*4+0], z = coords[i*4+1], y = coords[i*4+2], x = coords[i*4+3];
    int lin = ((b*DD + z)*HH + y)*WW + x;
    grid[lin] = i;
}

// ------------------------------------------- weight -> WMMA-B operand layout
// wT[(((t*16 + kc)*4 + nt)*32 + lane)*2 + {0,1}]
//   lane 0-15 : n = lane,    k = kc*4 + {0,1}
//   lane16-31 : n = lane-16, k = kc*4 + {2,3}
// source: weight[o][kd][kh][kw][c] = w[(o*27 + t)*64 + c],  o = nt*16 + n
__global__ void k_wt(const float* __restrict__ w, float* __restrict__ wT) {
    int i = blockIdx.x * blockDim.x + threadIdx.x;
    if (i >= NOFF*16*4*32) return;
    int lane = i & 31;
    int nt   = (i >> 5) & 3;
    int kc   = (i >> 7) & 15;
    int t    =  i >> 11;
    int o = nt*16 + (lane & 15);
    int c = kc*4 + 2*(lane >> 4);
    const float* src = w + (size_t)(o*NOFF + t)*CI + c;
    wT[(size_t)i*2 + 0] = src[0];
    wT[(size_t)i*2 + 1] = src[1];
}

// ------------------------------------------------------------------- conv
// block = 128 threads = 4 waves; block handles 16 points x 64 out channels;
// wave w handles output channels [16w, 16w+16).
__global__ __launch_bounds__(128)
void k_conv(const float* __restrict__ feats,
            const float* __restrict__ wT,
            const float* __restrict__ bias,
            const int*   __restrict__ coords,
            const int*   __restrict__ grid,
            float*       __restrict__ out) {
    __shared__ int   nbr[NOFF*16];
    __shared__ float A[16*A_STRIDE];

    const int tid  = threadIdx.x;
    const int lane = tid & 31;
    const int wv   = tid >> 5;        // N-tile (output-channel tile)
    const int p0   = blockIdx.x * 16; // first point of this tile (N % 16 == 0)

    // ---- 27x16 neighbor table via dense grid hash
    for (int i = tid; i < NOFF*16; i += 128) {
        int t = i >> 4, m = i & 15;
        int p = p0 + m;
        int bz = coords[p*4+0], z = coords[p*4+1], y = coords[p*4+2], x = coords[p*4+3];
        int kd = t / 9, kh = (t / 3) % 3, kw = t % 3;
        int nz = z + kd - 1, ny = y + kh - 1, nx = x + kw - 1;
        int idx = -1;
        if ((unsigned)nz < (unsigned)DD && (unsigned)ny < (unsigned)HH &&
            (unsigned)nx < (unsigned)WW) {
            int lin = ((bz*DD + nz)*HH + ny)*WW + nx;
            idx = grid[lin];
        }
        nbr[i] = idx;
    }
    __syncthreads();

    // ---- accumulators in WMMA C/D layout, seeded with bias (depends on n only)
    const int col  = lane & 15;   // = M for A-reads, = N for B/acc/store
    const int half = lane >> 4;
    float bv = bias[wv*16 + col];
    v8f acc;
#pragma unroll
    for (int j = 0; j < 8; ++j) acc[j] = bv;

    // A staging assignment: 8 threads per row, 8 floats (2x float4) each
    const int arow  = tid >> 3;
    const int acol8 = (tid & 7) * 8;

    for (int t = 0; t < NOFF; ++t) {
        // ---- gather 16x64 A tile (neighbor feature rows) into LDS
        int idx = nbr[t*16 + arow];
        float4 lo = make_float4(0.f, 0.f, 0.f, 0.f), hi = lo;
        if (idx >= 0) {
            const float4* src = (const float4*)(feats + (size_t)idx*CI + acol8);
            lo = src[0];
            hi = src[1];
        }
        float* dst = &A[arow*A_STRIDE + acol8];
        *(float4*)(dst)     = lo;
        *(float4*)(dst + 4) = hi;
        __syncthreads();

        // ---- 16 K-chunks of 4 over CI=64: v_wmma_f32_16x16x4_f32 chain
        const float* wbase = wT + (size_t)((t*16)*4 + wv)*64 + lane*2;
#pragma unroll
        for (int kc = 0; kc < 16; ++kc) {
            const float* ap = &A[col*A_STRIDE + kc*4 + 2*half];
            v2f a; a.x = ap[0]; a.y = ap[1];
            const float* bp = wbase + (size_t)kc*256;   // (kc*4)*64 floats
            v2f b; b.x = bp[0]; b.y = bp[1];
            acc = __builtin_amdgcn_wmma_f32_16x16x4_f32(
                false, a, false, b, (short)0, acc, false, false);
        }
        __syncthreads();
    }

    // ---- store D: VGPR j holds rows {j, j+8}, n = col
#pragma unroll
    for (int j = 0; j < 8; ++j) {
        int m = j + 8*half;
        out[(size_t)(p0 + m)*CO + wv*16 + col] = acc[j];
    }
}

// ---------------------------------------------------------------- launcher
extern "C" void kernel_launch(void* const* d_in, const int* in_sizes, int n_in,
                              void* d_out, int out_size, void* d_ws, size_t ws_size,
                              hipStream_t stream) {
    const float* feats  = (const float*)d_in[0];   // N x 64
    const float* weight = (const float*)d_in[1];   // 64 x 3 x 3 x 3 x 64
    const float* bias   = (const float*)d_in[2];   // 64
    const int*   coords = (const int*)  d_in[3];   // N x 4
    float* out = (float*)d_out;

    const int n = in_sizes[0] / CI;                // 200000

    float* wT   = (float*)d_ws;                               // 442,368 B
    int*   grid = (int*)((char*)d_ws + (size_t)WT_FLOATS*4);  // 16,777,216 B (256B aligned)

    k_fill_grid<<<(GRID_CELLS + 255)/256, 256, 0, stream>>>(grid);
    k_scatter  <<<(n + 255)/256,          256, 0, stream>>>(coords, grid, n);
    k_wt       <<<(NOFF*16*4*32 + 255)/256, 256, 0, stream>>>(weight, wT);
    k_conv     <<<n/16, 128, 0, stream>>>(feats, wT, bias, coords, grid, out);
}